// SpatialEquivariantAttention_65635690217683
// MI455X (gfx1250) — compile-verified
//
#include <hip/hip_runtime.h>
#include <hip/hip_bf16.h>

// ---------------------------------------------------------------------------
// SpatialEquivariantAttention for MI455X (gfx1250, wave32, WMMA f32 16x16x4)
// ---------------------------------------------------------------------------

#define B_  2
#define S_  2048
#define D_  64
#define H_  32
#define LDW 66   // padded LDS row stride (floats) for 64-wide tiles
#define LDV 18   // padded LDS row stride (floats) for 16-wide tiles

typedef float v2f __attribute__((ext_vector_type(2)));
typedef float v8f __attribute__((ext_vector_type(8)));

// ---------------------------------------------------------------------------
// Kernel 1: fused QKV projection.  Q/K/V[b,s,:] = x[b,s,:] @ W{q,k,v}
// One block = one 16-row tile of x; 4 waves each own a 16-wide d-tile and
// loop over the three weight matrices.  All GEMM work on v_wmma_f32_16x16x4.
// ---------------------------------------------------------------------------
__global__ __launch_bounds__(128) void qkv_kernel(
    const float* __restrict__ x,
    const float* __restrict__ Wq, const float* __restrict__ Wk,
    const float* __restrict__ Wv,
    float* __restrict__ Q, float* __restrict__ K, float* __restrict__ V) {
  __shared__ float xt[16 * LDW];       // x tile, row major [m][k]
  __shared__ float wt[3][64 * LDW];    // W transposed: [n][k]

  const int tid  = threadIdx.x;
  const int blk  = blockIdx.x;                 // 0 .. B*S/16-1
  const int b    = blk / (S_ / 16);
  const int i0   = (blk % (S_ / 16)) * 16;

  const float* xrow = x + ((size_t)b * S_ + i0) * D_;
  for (int idx = tid; idx < 16 * 64; idx += 128) {
    xt[(idx >> 6) * LDW + (idx & 63)] = xrow[idx];
  }
  const float* Ws[3] = {Wq, Wk, Wv};
  for (int m = 0; m < 3; ++m) {
    for (int idx = tid; idx < 64 * 64; idx += 128) {
      const int r = idx >> 6, c = idx & 63;    // W[k=r][n=c]
      wt[m][c * LDW + r] = Ws[m][idx];         // transposed store
    }
  }
  __syncthreads();

  const int lane = tid & 31;
  const int wave = tid >> 5;
  const int half = lane >> 4;     // 0 | 1
  const int l16  = lane & 15;
  const int n0   = wave * 16;     // this wave's output d-tile

  float* outs[3] = {Q, K, V};
  for (int m = 0; m < 3; ++m) {
    v8f c = {0, 0, 0, 0, 0, 0, 0, 0};
#pragma unroll
    for (int kk = 0; kk < 16; ++kk) {
      const int ko = kk * 4 + 2 * half;
      v2f a  = *(const v2f*)&xt[l16 * LDW + ko];
      v2f bb = *(const v2f*)&wt[m][(n0 + l16) * LDW + ko];
      c = __builtin_amdgcn_wmma_f32_16x16x4_f32(false, a, false, bb,
                                                (short)0, c, false, false);
    }
    float* op = outs[m] + ((size_t)b * S_ + i0) * D_;
#pragma unroll
    for (int r = 0; r < 8; ++r) {
      op[(r + 8 * half) * D_ + n0 + l16] = c[r];
    }
  }
}

// ---------------------------------------------------------------------------
// Kernel 2: flash attention with pairwise-MLP score bias.
// One block = one 16-query tile.  4 waves split the 128 key tiles
// (online softmax per wave), merged at the end through LDS.
// psi terms: sin/cos via rsqrt identity, refl folded to 2-input weights,
// all per-h weights packed into one 64B LDS line.
// ---------------------------------------------------------------------------
__global__ __launch_bounds__(128) void attn_kernel(
    const float* __restrict__ Q, const float* __restrict__ K,
    const float* __restrict__ V, const float* __restrict__ coords,
    const float* __restrict__ rot_w1, const float* __restrict__ rot_b1,
    const float* __restrict__ rot_w2, const float* __restrict__ rot_b2,
    const float* __restrict__ trans_w1, const float* __restrict__ trans_b1,
    const float* __restrict__ trans_w2, const float* __restrict__ trans_b2,
    const float* __restrict__ refl_w1, const float* __restrict__ refl_b1,
    const float* __restrict__ refl_w2, const float* __restrict__ refl_b2,
    float* __restrict__ out) {
  __shared__ float qt[16 * LDW];                       // Q tile [m][k]
  __shared__ float kt[4][16 * LDW];                    // K tile per wave [n][k]
  __shared__ float vt[4][64 * LDV];                    // V tile transposed [d][k]
  __shared__ float pst[4][16 * LDV];                   // P staging [m][n]
  __shared__ __align__(16) float wpack[H_][16];        // packed per-h weights
  __shared__ float qc[32];                             // query coords (16 x,y)
  __shared__ float kc[4][32];                          // key coords per wave
  __shared__ float red_m[4][16], red_l[4][16];
  __shared__ float red_acc[4][16 * 64];

  const int tid  = threadIdx.x;
  const int lane = tid & 31;
  const int wave = tid >> 5;
  const int half = lane >> 4;
  const int l16  = lane & 15;
  const int blk  = blockIdx.x;
  const int b    = blk / (S_ / 16);
  const int i0   = (blk % (S_ / 16)) * 16;

  // ---- stage query tile, query coords, packed MLP weights -----------------
  const float* qrow = Q + ((size_t)b * S_ + i0) * D_;
  for (int idx = tid; idx < 16 * 64; idx += 128) {
    qt[(idx >> 6) * LDW + (idx & 63)] = qrow[idx];
  }
  if (tid < 32) qc[tid] = coords[((size_t)b * S_ + i0) * 2 + tid];
  if (tid < H_) {
    const int h = tid;
    float* wp = &wpack[h][0];
    wp[0]  = rot_w1[h];                    // rot w1 row0
    wp[1]  = rot_w1[H_ + h];               // rot w1 row1
    wp[2]  = rot_w1[2 * H_ + h];           // rot w1 row2
    wp[3]  = rot_b1[h];
    wp[4]  = rot_w2[h];
    wp[5]  = trans_w1[h];
    wp[6]  = trans_w1[H_ + h];
    wp[7]  = trans_b1[h];
    wp[8]  = trans_w2[h];
    wp[9]  = refl_w1[h]      - refl_w1[2 * H_ + h];  // folded reflection
    wp[10] = refl_w1[H_ + h] - refl_w1[3 * H_ + h];
    wp[11] = refl_b1[h];
    wp[12] = refl_w2[h];
    wp[13] = 0.f; wp[14] = 0.f; wp[15] = 0.f;
  }
  __syncthreads();

  const float psi_c = rot_b2[0] + trans_b2[0] + refl_b2[0];

  float cix[8], ciy[8];
#pragma unroll
  for (int r = 0; r < 8; ++r) {
    const int m = r + 8 * half;
    cix[r] = qc[2 * m];
    ciy[r] = qc[2 * m + 1];
  }

  float m_row[8], l_row[8], alph[8];
  v8f acc[4];
#pragma unroll
  for (int r = 0; r < 8; ++r) { m_row[r] = -1e30f; l_row[r] = 0.f; }
#pragma unroll
  for (int dt = 0; dt < 4; ++dt) acc[dt] = (v8f){0, 0, 0, 0, 0, 0, 0, 0};

  const int NKT = S_ / 16;   // 128 key tiles, split 4 ways -> 32 iters/wave
  for (int kti = wave; kti < NKT; kti += 4) {
    const int j0 = kti * 16;
    // ---- stage K row-major, V transposed, key coords (this wave only) -----
    const float* krow = K + ((size_t)b * S_ + j0) * D_;
    const float* vrow = V + ((size_t)b * S_ + j0) * D_;
    for (int idx = lane; idx < 16 * 64; idx += 32) {
      const int r = idx >> 6, c = idx & 63;
      kt[wave][r * LDW + c] = krow[idx];
      vt[wave][c * LDV + r] = vrow[idx];
    }
    kc[wave][lane] = coords[((size_t)b * S_ + j0) * 2 + lane];
    __syncthreads();   // uniform: every wave does exactly NKT/4 iterations

    // ---- scores tile = Q Kt^T via 16 x v_wmma_f32_16x16x4_f32 -------------
    v8f c = {0, 0, 0, 0, 0, 0, 0, 0};
#pragma unroll
    for (int kk = 0; kk < 16; ++kk) {
      const int ko = kk * 4 + 2 * half;
      v2f a  = *(const v2f*)&qt[l16 * LDW + ko];
      v2f bb = *(const v2f*)&kt[wave][l16 * LDW + ko];
      c = __builtin_amdgcn_wmma_f32_16x16x4_f32(false, a, false, bb,
                                                (short)0, c, false, false);
    }

    // ---- per-pair geometry (rsqrt identity replaces atan2/sin/cos) --------
    const float cjx = kc[wave][2 * l16];
    const float cjy = kc[wave][2 * l16 + 1];
    float dxv[8], dyv[8], dsv[8], snv[8], csv[8];
#pragma unroll
    for (int r = 0; r < 8; ++r) {
      const float dx = cjx - cix[r];
      const float dy = cjy - ciy[r];
      const float d2 = fmaf(dx, dx, fmaf(dy, dy, 1e-8f));
      const float ri = rsqrtf(d2);
      float cs = dx * ri;
      if (dx == 0.f && dy == 0.f) cs = 1.f;   // atan2(0,0)=0 -> cos=1
      dxv[r] = dx; dyv[r] = dy;
      dsv[r] = d2 * ri;                       // sqrt(d2)
      snv[r] = dy * ri;
      csv[r] = cs;
    }

    // ---- three MLP biases, 13 VALU ops per (pair, h) ----------------------
    float psi[8];
#pragma unroll
    for (int r = 0; r < 8; ++r) psi[r] = psi_c;
#pragma unroll 4
    for (int h = 0; h < H_; ++h) {
      const float4* wp4 = (const float4*)&wpack[h][0];
      const float4 wa = wp4[0];   // rot w1[0..2], rot b1
      const float4 wb = wp4[1];   // rot w2, trans w1[0..1], trans b1
      const float4 wc = wp4[2];   // trans w2, refl f0, f1, refl b1
      const float4 wd = wp4[3];   // refl w2
#pragma unroll
      for (int r = 0; r < 8; ++r) {
        const float h1 = fmaf(dsv[r], wa.x,
                         fmaf(snv[r], wa.y, fmaf(csv[r], wa.z, wa.w)));
        const float h2 = fmaf(dxv[r], wb.y, fmaf(dyv[r], wb.z, wb.w));
        const float h3 = fmaf(dxv[r], wc.y, fmaf(dyv[r], wc.z, wc.w));
        psi[r] = fmaf(fmaxf(h1, 0.f), wb.x, psi[r]);
        psi[r] = fmaf(fmaxf(h2, 0.f), wc.x, psi[r]);
        psi[r] = fmaf(fmaxf(h3, 0.f), wd.x, psi[r]);
      }
    }

    // ---- online softmax ---------------------------------------------------
    float sv[8];
#pragma unroll
    for (int r = 0; r < 8; ++r) sv[r] = fmaf(c[r], 0.125f, psi[r]);
#pragma unroll
    for (int r = 0; r < 8; ++r) {
      float mx = sv[r];
      mx = fmaxf(mx, __shfl_xor(mx, 1, 32));
      mx = fmaxf(mx, __shfl_xor(mx, 2, 32));
      mx = fmaxf(mx, __shfl_xor(mx, 4, 32));
      mx = fmaxf(mx, __shfl_xor(mx, 8, 32));
      const float m_new = fmaxf(m_row[r], mx);
      const float al = __expf(m_row[r] - m_new);
      float p = __expf(sv[r] - m_new);
      float rs = p;
      rs += __shfl_xor(rs, 1, 32);
      rs += __shfl_xor(rs, 2, 32);
      rs += __shfl_xor(rs, 4, 32);
      rs += __shfl_xor(rs, 8, 32);
      l_row[r] = fmaf(l_row[r], al, rs);
      m_row[r] = m_new;
      alph[r] = al;
      sv[r] = p;
    }
#pragma unroll
    for (int dt = 0; dt < 4; ++dt)
#pragma unroll
      for (int r = 0; r < 8; ++r) acc[dt][r] *= alph[r];

    // ---- restage P (C-layout -> A-layout via LDS), then P @ V -------------
#pragma unroll
    for (int r = 0; r < 8; ++r)
      pst[wave][(r + 8 * half) * LDV + l16] = sv[r];
    __syncthreads();   // uniform trip count across waves

#pragma unroll
    for (int dt = 0; dt < 4; ++dt) {
      const int d0 = dt * 16;
#pragma unroll
      for (int kk = 0; kk < 4; ++kk) {
        const int ko = kk * 4 + 2 * half;
        v2f a  = *(const v2f*)&pst[wave][l16 * LDV + ko];
        v2f bb = *(const v2f*)&vt[wave][(d0 + l16) * LDV + ko];
        acc[dt] = __builtin_amdgcn_wmma_f32_16x16x4_f32(
            false, a, false, bb, (short)0, acc[dt], false, false);
      }
    }
  }

  // ---- merge the 4 waves' partial softmax states --------------------------
  if (l16 == 0) {
#pragma unroll
    for (int r = 0; r < 8; ++r) {
      red_m[wave][r + 8 * half] = m_row[r];
      red_l[wave][r + 8 * half] = l_row[r];
    }
  }
#pragma unroll
  for (int dt = 0; dt < 4; ++dt)
#pragma unroll
    for (int r = 0; r < 8; ++r)
      red_acc[wave][(r + 8 * half) * 64 + dt * 16 + l16] = acc[dt][r];
  __syncthreads();

  float* orow = out + ((size_t)b * S_ + i0) * D_;
  for (int idx = tid; idx < 16 * 64; idx += 128) {
    const int m = idx >> 6, d = idx & 63;
    const float M = fmaxf(fmaxf(red_m[0][m], red_m[1][m]),
                          fmaxf(red_m[2][m], red_m[3][m]));
    float Ltot = 0.f, val = 0.f;
#pragma unroll
    for (int w = 0; w < 4; ++w) {
      const float e = __expf(red_m[w][m] - M);
      Ltot = fmaf(red_l[w][m], e, Ltot);
      val  = fmaf(red_acc[w][m * 64 + d], e, val);
    }
    orow[idx] = val / Ltot;
  }
}

// ---------------------------------------------------------------------------
extern "C" void kernel_launch(void* const* d_in, const int* in_sizes, int n_in,
                              void* d_out, int out_size, void* d_ws,
                              size_t ws_size, hipStream_t stream) {
  const float* x        = (const float*)d_in[0];
  const float* coords   = (const float*)d_in[1];
  const float* Wq       = (const float*)d_in[2];
  const float* Wk       = (const float*)d_in[3];
  const float* Wv       = (const float*)d_in[4];
  const float* rot_w1   = (const float*)d_in[5];
  const float* rot_b1   = (const float*)d_in[6];
  const float* rot_w2   = (const float*)d_in[7];
  const float* rot_b2   = (const float*)d_in[8];
  const float* trans_w1 = (const float*)d_in[9];
  const float* trans_b1 = (const float*)d_in[10];
  const float* trans_w2 = (const float*)d_in[11];
  const float* trans_b2 = (const float*)d_in[12];
  const float* refl_w1  = (const float*)d_in[13];
  const float* refl_b1  = (const float*)d_in[14];
  const float* refl_w2  = (const float*)d_in[15];
  const float* refl_b2  = (const float*)d_in[16];
  float* outp = (float*)d_out;

  const size_t N = (size_t)B_ * S_ * D_;   // 262144 floats
  float* Qw = (float*)d_ws;
  float* Kw = Qw + N;
  float* Vw = Kw + N;

  const int nblk = B_ * (S_ / 16);         // 256
  qkv_kernel<<<nblk, 128, 0, stream>>>(x, Wq, Wk, Wv, Qw, Kw, Vw);
  attn_kernel<<<nblk, 128, 0, stream>>>(Qw, Kw, Vw, coords,
                                        rot_w1, rot_b1, rot_w2, rot_b2,
                                        trans_w1, trans_b1, trans_w2, trans_b2,
                                        refl_w1, refl_b1, refl_w2, refl_b2,
                                        outp);
}